// BoxLoss_23381801959719
// MI455X (gfx1250) — compile-verified
//
#include <hip/hip_runtime.h>
#include <math.h>

// CIoU box loss for MI455X (gfx1250, wave32).
//
// Roofline: 159 MB streamed (2x 68.8 MB f32x4 boxes + 17.2 MB box_norm +
// 4.3 MB mask) -> ~6.8 us HBM floor at 23.3 TB/s; working set < 192 MB L2 so
// graph replays can run L2-resident (default RT temporal hints on purpose).
// Per-element VALU (~80 ops incl. two atan polynomials) is the same order,
// so: vector b128 loads, lean fp32 math, x2 unrolled grid-stride loop.
//
// Reductions:
//   * wave: V_WMMA_F32_16X16X4_F32 with A = ones folds lanes n and n+16 into
//     16 fp32 column sums in one matrix op, then 4 one-instruction
//     ds_swizzle_b32 xor steps (group-of-32 mode) finish the wave sum.
//   * block: 8 wave sums via LDS; grid: per-block partials in d_ws, one-block
//     finalize divides by cls_norm. No atomics -> bitwise-deterministic.

typedef float v2f __attribute__((ext_vector_type(2)));
typedef float v8f __attribute__((ext_vector_type(8)));

#define CIOU_EPS       1e-9f
#define CIOU_PI2_INV4  0.405284734569351086f  // 4 / pi^2

// One-instruction xor-shuffle-add: ds_swizzle_b32 group-of-32 mode,
// offset = {0, xor_mask[14:10], or_mask=0, and_mask=0x1f}.
template <int XORM>
__device__ __forceinline__ float xor_swizzle_add(float v) {
    const int sw = __builtin_amdgcn_ds_swizzle(__float_as_int(v),
                                               (XORM << 10) | 0x1f);
    return v + __int_as_float(sw);
}

// Reduce one fp32 partial per lane to a full-wave sum using the matrix unit.
// B matrix (4x16 f32, 2 VGPRs): VGPR0 = lane partial, VGPR1 = 0. With A = ones
// (16x4), D[m,n] = colsum_n = p[lane n] + p[lane n+16] for every m. Lanes
// 16..31 hold duplicate column sums but reduce within their own half, so all
// lanes end with the same wave total. EXEC must be all-ones (called after full
// reconvergence, full 256-thread blocks).
__device__ __forceinline__ float wave_sum_wmma(float p) {
    v2f ones; ones[0] = 1.0f; ones[1] = 1.0f;
    v2f bmat; bmat[0] = p;    bmat[1] = 0.0f;
    v8f c = {};
    c = __builtin_amdgcn_wmma_f32_16x16x4_f32(
        /*neg_a=*/false, ones, /*neg_b=*/false, bmat,
        /*c_mod=*/(short)0, c, /*reuse_a=*/false, /*reuse_b=*/false);
    float s = c[0];
    s = xor_swizzle_add<1>(s);
    s = xor_swizzle_add<2>(s);
    s = xor_swizzle_add<4>(s);
    s = xor_swizzle_add<8>(s);
    return s;
}

__device__ __forceinline__ float ciou_contrib(const float4* __restrict__ pb,
                                              const float4* __restrict__ tb,
                                              const unsigned char* __restrict__ mask,
                                              const float* __restrict__ bn,
                                              int idx) {
    const float4 a = pb[idx];   // global_load_b128
    const float4 b = tb[idx];   // global_load_b128

    // intersection
    float iw = fminf(a.z, b.z) - fmaxf(a.x, b.x);
    float ih = fminf(a.w, b.w) - fmaxf(a.y, b.y);
    iw = fmaxf(iw, 0.0f);
    ih = fmaxf(ih, 0.0f);
    const float inter = iw * ih;

    // union / IoU
    const float wa = a.z - a.x, ha = a.w - a.y;
    const float wb = b.z - b.x, hb = b.w - b.y;
    const float uni = wa * ha + wb * hb - inter;
    const float iou = inter / (uni + CIOU_EPS);

    // center distance over enclosing diagonal
    const float dx = (a.x + a.z) - (b.x + b.z);
    const float dy = (a.y + a.w) - (b.y + b.w);
    const float cent = 0.25f * (dx * dx + dy * dy);
    const float cw = fmaxf(a.z, b.z) - fminf(a.x, b.x);
    const float ch = fmaxf(a.w, b.w) - fminf(a.y, b.y);
    const float diag = cw * cw + ch * ch;
    const float diou = iou - cent / (diag + CIOU_EPS);

    // aspect-ratio consistency
    const float d  = atanf(wa / (ha + CIOU_EPS)) - atanf(wb / (hb + CIOU_EPS));
    const float v  = CIOU_PI2_INV4 * d * d;
    const float al = v / (v - iou + 1.0f + CIOU_EPS);
    const float ciou = diou - al * v;

    const float li = 1.0f - fminf(fmaxf(ciou, 0.0f), 1.0f);
    const float m  = mask[idx] ? 1.0f : 0.0f;
    return li * bn[idx] * m;
}

__global__ void __launch_bounds__(256)
ciou_partial_kernel(const float4* __restrict__ pb,
                    const float4* __restrict__ tb,
                    const unsigned char* __restrict__ mask,
                    const float* __restrict__ bn,
                    float* __restrict__ partials,
                    int n) {
    const int tid    = threadIdx.x;
    const int stride = (int)(gridDim.x * blockDim.x);
    const int start  = (int)(blockIdx.x * blockDim.x) + tid;

    float acc = 0.0f;
    // x2 unrolled grid-stride loop: two independent b128 load pairs in flight.
    for (int idx = start; idx < n; idx += 2 * stride) {
        float p0 = ciou_contrib(pb, tb, mask, bn, idx);
        float p1 = 0.0f;
        const int idx1 = idx + stride;
        if (idx1 < n) p1 = ciou_contrib(pb, tb, mask, bn, idx1);
        acc += p0 + p1;
    }

    // EXEC all-ones here (fully reconverged, full 256-thread blocks).
    const float wsum = wave_sum_wmma(acc);

    __shared__ float sm[8];
    const int wave = tid >> 5;
    if ((tid & 31) == 0) sm[wave] = wsum;
    __syncthreads();
    if (tid == 0) {
        float s = 0.0f;
#pragma unroll
        for (int i = 0; i < 8; ++i) s += sm[i];
        partials[blockIdx.x] = s;
    }
}

__global__ void __launch_bounds__(256)
ciou_finalize_kernel(const float* __restrict__ partials, int nparts,
                     const float* __restrict__ cls_norm,
                     float* __restrict__ out) {
    const int tid = threadIdx.x;
    float s = 0.0f;
    for (int i = tid; i < nparts; i += (int)blockDim.x) s += partials[i];

    s = xor_swizzle_add<16>(s);
    s = xor_swizzle_add<8>(s);
    s = xor_swizzle_add<4>(s);
    s = xor_swizzle_add<2>(s);
    s = xor_swizzle_add<1>(s);

    __shared__ float sm[8];
    const int wave = tid >> 5;
    if ((tid & 31) == 0) sm[wave] = s;
    __syncthreads();
    if (tid == 0) {
        float t = 0.0f;
#pragma unroll
        for (int i = 0; i < 8; ++i) t += sm[i];
        out[0] = t / cls_norm[0];
    }
}

extern "C" void kernel_launch(void* const* d_in, const int* in_sizes, int n_in,
                              void* d_out, int out_size, void* d_ws, size_t ws_size,
                              hipStream_t stream) {
    // setup_inputs() order:
    //   0: predicts_bbox [B, A, 4] f32
    //   1: targets_bbox  [B, A, 4] f32
    //   2: valid_masks   [B, A]    bool (1 byte/elem)
    //   3: box_norm      [B, A]    f32
    //   4: cls_norm      scalar    f32
    const float4*        pb   = (const float4*)d_in[0];
    const float4*        tb   = (const float4*)d_in[1];
    const unsigned char* msk  = (const unsigned char*)d_in[2];
    const float*         bn   = (const float*)d_in[3];
    const float*         cls  = (const float*)d_in[4];
    float*               out  = (float*)d_out;
    float*               part = (float*)d_ws;

    const int n = in_sizes[3];  // B*A elements

    int nblocks = 2048;  // 2048 blocks x 8 waves -> deep latency hiding
    const size_t maxp = ws_size / sizeof(float);
    if ((size_t)nblocks > maxp) nblocks = (int)maxp;
    if (nblocks < 1) nblocks = 1;

    ciou_partial_kernel<<<dim3((unsigned)nblocks), dim3(256), 0, stream>>>(
        pb, tb, msk, bn, part, n);
    ciou_finalize_kernel<<<dim3(1), dim3(256), 0, stream>>>(part, nblocks, cls, out);
}